// PointNet2MSGClassifier_8478265442718
// MI455X (gfx1250) — compile-verified
//
#include <hip/hip_runtime.h>
#include <hip/hip_bf16.h>

typedef __attribute__((ext_vector_type(2))) float v2f;
typedef __attribute__((ext_vector_type(8))) float v8f;

#define EPSBN 1e-5f

// =====================================================================
// GEMM via V_WMMA_F32_16X16X4_F32 (fp32 matrix core, matches reference
// precision).  Out[M x N] = W[M x K] * F[K x N], all row-major.
// K must be a multiple of 4 (feat buffer / weights are pre-padded).
// grid = (N/256, M/16), block = 128 (4 waves); wave w owns cols
// [256*bx + 64*w, +64).  Inner loop is branchless: A is one b64 load,
// B is 8 coalesced b32 loads, 4 WMMAs per K-step, prefetch ahead on F.
// A layout (16x4 f32): lane L: a[v] = A[L%16][2*(L/16)+v]
// B layout (4x16 f32): b[v] = B[2*(L/16)+v][L%16]
// C/D layout (16x16):  acc[i] = C[i + 8*(L/16)][L%16]
// =====================================================================
__global__ __launch_bounds__(128) void gemm_wmma_f32(
    const float* __restrict__ W, const float* __restrict__ F,
    float* __restrict__ Out, int K, int Ncols)
{
  const int lane = threadIdx.x & 31;
  const int wid  = threadIdx.x >> 5;
  const int half = lane >> 4;
  const int mn   = lane & 15;
  const int row0 = blockIdx.y << 4;
  const int col0 = (blockIdx.x << 8) + (wid << 6);

  v8f acc0 = {}; v8f acc1 = {}; v8f acc2 = {}; v8f acc3 = {};
  const float*  Wr      = W + (size_t)(row0 + mn) * (size_t)K + 2 * half;
  const float*  Fc      = F + (size_t)(2 * half) * (size_t)Ncols + col0 + mn;
  const size_t  rowstep = (size_t)Ncols;

  #pragma unroll 2
  for (int k0 = 0; k0 < K; k0 += 4) {
    const v2f a = *(const v2f*)Wr;
    const float* F0 = Fc;
    const float* F1 = Fc + rowstep;
    v2f b0, b1, b2, b3;
    b0.x = F0[0];  b0.y = F1[0];
    b1.x = F0[16]; b1.y = F1[16];
    b2.x = F0[32]; b2.y = F1[32];
    b3.x = F0[48]; b3.y = F1[48];
    __builtin_prefetch(Fc + 8 * rowstep, 0, 1);   // global_prefetch on F stream
    acc0 = __builtin_amdgcn_wmma_f32_16x16x4_f32(false, a, false, b0, (short)0, acc0, false, false);
    acc1 = __builtin_amdgcn_wmma_f32_16x16x4_f32(false, a, false, b1, (short)0, acc1, false, false);
    acc2 = __builtin_amdgcn_wmma_f32_16x16x4_f32(false, a, false, b2, (short)0, acc2, false, false);
    acc3 = __builtin_amdgcn_wmma_f32_16x16x4_f32(false, a, false, b3, (short)0, acc3, false, false);
    Wr += 4;
    Fc += 4 * rowstep;
  }
  #pragma unroll
  for (int i = 0; i < 8; ++i) {
    const int r = row0 + i + 8 * half;
    float* o = Out + (size_t)r * (size_t)Ncols + col0 + mn;
    o[0]  = acc0[i];
    o[16] = acc1[i];
    o[32] = acc2[i];
    o[48] = acc3[i];
  }
}

// Zero-pad first-layer weights along K: Wp[M x Kp] from W[M x K]
__global__ void pad_weights(const float* __restrict__ W, float* __restrict__ Wp,
                            int M, int K, int Kp)
{
  const int i = blockIdx.x * blockDim.x + threadIdx.x;
  if (i >= M * Kp) return;
  const int k = i % Kp;
  const int o = i / Kp;
  Wp[i] = (k < K) ? W[(size_t)o * K + k] : 0.0f;
}

// =====================================================================
// k-NN top-k: one block per (b,s) query.  Distances via the reference's
// |a|^2 + |b|^2 - 2 a.b formula; k rounds of parallel argmin with
// lowest-index tie-break (matches lax.top_k ordering on -d2).
// =====================================================================
#define TKT 256
__global__ __launch_bounds__(TKT) void knn_topk(
    const float* __restrict__ xyz, int B, int N, int S, int Kk,
    int* __restrict__ idx)
{
  __shared__ float dist[4096];
  __shared__ float rmin[TKT];
  __shared__ int   rarg[TKT];
  const int bs = blockIdx.x;
  const int b  = bs / S;
  const int s  = bs - b * S;
  const float* xb = xyz + (size_t)b * 3 * N;
  const float qx = xb[s], qy = xb[N + s], qz = xb[2 * N + s];
  const float qq = qx * qx + qy * qy + qz * qz;
  const int t = threadIdx.x;

  for (int n = t; n < N; n += TKT) {
    const float px = xb[n], py = xb[N + n], pz = xb[2 * N + n];
    const float nn = px * px + py * py + pz * pz;
    const float dd = qx * px + qy * py + qz * pz;
    dist[n] = qq + nn - 2.0f * dd;
  }
  __syncthreads();

  int* out = idx + (size_t)bs * Kk;
  for (int j = 0; j < Kk; ++j) {
    float best = 3.4e38f; int barg = N;
    for (int n = t; n < N; n += TKT) {
      const float d = dist[n];
      if (d < best) { best = d; barg = n; }
    }
    rmin[t] = best; rarg[t] = barg;
    __syncthreads();
    for (int w = TKT / 2; w > 0; w >>= 1) {
      if (t < w) {
        const float o = rmin[t + w]; const int oa = rarg[t + w];
        if (o < rmin[t] || (o == rmin[t] && oa < rarg[t])) { rmin[t] = o; rarg[t] = oa; }
      }
      __syncthreads();
    }
    if (t == 0) { out[j] = rarg[0]; dist[rarg[0]] = 3.4e38f; }
    __syncthreads();
  }
}

// =====================================================================
// Build feature matrix F[Kp x (B*S*K)]: rows 0..2 = gathered xyz minus
// query, rows 3..(3+C-1) = gathered point features, rows (3+C)..Kp-1 = 0
// (K padding so the WMMA GEMM needs no edge guards).
// =====================================================================
__global__ void build_feat(
    const float* __restrict__ xyz, const float* __restrict__ pts,
    const int* __restrict__ idx, float* __restrict__ F,
    int B, int N, int S, int Kk, int C, int Kp)
{
  const size_t cols = (size_t)B * S * Kk;
  const size_t col  = (size_t)blockIdx.x * blockDim.x + threadIdx.x;
  if (col >= cols) return;
  const size_t bs = col / Kk;
  const int b = (int)(bs / S);
  const int s = (int)(bs - (size_t)b * S);
  const int j = idx[col];
  const float* xb = xyz + (size_t)b * 3 * N;
  F[col]            = xb[j]         - xb[s];
  F[cols + col]     = xb[N + j]     - xb[N + s];
  F[2 * cols + col] = xb[2 * N + j] - xb[2 * N + s];
  if (pts) {
    const float* pb = pts + (size_t)b * C * N;
    for (int c = 0; c < C; ++c)
      F[(size_t)(3 + c) * cols + col] = pb[(size_t)c * N + j];
  }
  for (int c = 3 + C; c < Kp; ++c)
    F[(size_t)c * cols + col] = 0.0f;
}

// =====================================================================
// Batch-norm statistics over the column axis (batch+spatial folded in):
// one block per channel.  Produces fused scale/shift.  (Layer biases
// cancel inside BN, so GEMM output needs no bias.)
// =====================================================================
__global__ __launch_bounds__(256) void bn_stats(
    const float* __restrict__ Z, size_t cols,
    const float* __restrict__ gamma, const float* __restrict__ beta,
    float* __restrict__ scale, float* __restrict__ shift)
{
  __shared__ float s1[256], s2[256];
  const int o = blockIdx.x, t = threadIdx.x;
  const float* z = Z + (size_t)o * cols;
  float a = 0.0f, b = 0.0f;
  for (size_t i = t; i < cols; i += 256) { const float v = z[i]; a += v; b += v * v; }
  s1[t] = a; s2[t] = b; __syncthreads();
  for (int w = 128; w > 0; w >>= 1) {
    if (t < w) { s1[t] += s1[t + w]; s2[t] += s2[t + w]; }
    __syncthreads();
  }
  if (t == 0) {
    const float mean = s1[0] / (float)cols;
    const float var  = s2[0] / (float)cols - mean * mean;
    const float inv  = rsqrtf(var + EPSBN);
    const float sc   = gamma[o] * inv;
    scale[o] = sc;
    shift[o] = beta[o] - mean * sc;
  }
}

__global__ void bn_apply(float* __restrict__ Z, size_t cols,
                         const float* __restrict__ sc, const float* __restrict__ sh,
                         int relu)
{
  const int o = blockIdx.y;
  const size_t i = (size_t)blockIdx.x * blockDim.x + threadIdx.x;
  if (i >= cols) return;
  float v = Z[(size_t)o * cols + i] * sc[o] + sh[o];
  if (relu) v = fmaxf(v, 0.0f);
  Z[(size_t)o * cols + i] = v;
}

// Max over neighbor axis: act[Cout x (B*S*K)] -> P[b][coff+o][s]
__global__ void maxpool_k(const float* __restrict__ A, float* __restrict__ P,
                          int B, int S, int Kk, int Cout, int Ctot, int coff)
{
  const int i = blockIdx.x * blockDim.x + threadIdx.x;
  const int tot = B * Cout * S;
  if (i >= tot) return;
  const int s = i % S;
  const int r = i / S;
  const int o = r % Cout;
  const int b = r / Cout;
  const size_t cols = (size_t)B * S * Kk;
  const size_t base = (size_t)o * cols + (size_t)(b * S + s) * Kk;
  float m = -3.4e38f;
  for (int n = 0; n < Kk; ++n) m = fmaxf(m, A[base + n]);
  P[((size_t)b * Ctot + coff + o) * S + s] = m;
}

// Slice channel range [c0,c0+cnt) of X[B,Ctot,N] into Y[B,cnt,N]
__global__ void slice_ch(const float* __restrict__ X, float* __restrict__ Y,
                         int B, int Ctot, int N, int c0, int cnt)
{
  const int i = blockIdx.x * blockDim.x + threadIdx.x;
  if (i >= B * cnt * N) return;
  const int n = i % N;
  const int r = i / N;
  const int c = r % cnt;
  const int b = r / cnt;
  Y[((size_t)b * cnt + c) * N + n] = X[((size_t)b * Ctot + c0 + c) * N + n];
}

// Prefix along last dim: Y[B,C,S] = X[B,C,:S]
__global__ void slice_n(const float* __restrict__ X, float* __restrict__ Y,
                        int B, int C, int N, int S)
{
  const int i = blockIdx.x * blockDim.x + threadIdx.x;
  if (i >= B * C * S) return;
  const int s = i % S;
  const int r = i / S;
  const int c = r % C;
  const int b = r / C;
  Y[((size_t)b * C + c) * S + s] = X[((size_t)b * C + c) * N + s];
}

// Global max over last axis: H[b][c] = max_s P[b][c][s]
__global__ void max_over_s(const float* __restrict__ P, float* __restrict__ H,
                           int B, int C, int S)
{
  const int i = blockIdx.x * blockDim.x + threadIdx.x;
  if (i >= B * C) return;
  const float* p = P + (size_t)i * S;
  float m = -3.4e38f;
  for (int s = 0; s < S; ++s) m = fmaxf(m, p[s]);
  H[i] = m;
}

// Small FC: Y[b][o] = dot(W[o], H[b]) (+ bias)
__global__ void fc_dot(const float* __restrict__ H, const float* __restrict__ W,
                       const float* __restrict__ bias, float* __restrict__ Y,
                       int Bb, int Din, int Dout)
{
  const int i = blockIdx.x * blockDim.x + threadIdx.x;
  if (i >= Bb * Dout) return;
  const int o = i % Dout;
  const int b = i / Dout;
  const float* h = H + (size_t)b * Din;
  const float* w = W + (size_t)o * Din;
  float acc = bias ? bias[o] : 0.0f;
  for (int k = 0; k < Din; ++k) acc += h[k] * w[k];
  Y[(size_t)b * Dout + o] = acc;
}

// BN over the batch axis (8 samples) + ReLU, in place.
__global__ void bn_batch(float* __restrict__ Y, int Bb, int D,
                         const float* __restrict__ g, const float* __restrict__ be)
{
  const int o = blockIdx.x * blockDim.x + threadIdx.x;
  if (o >= D) return;
  float m = 0.0f;
  for (int b = 0; b < Bb; ++b) m += Y[(size_t)b * D + o];
  m /= (float)Bb;
  float v = 0.0f;
  for (int b = 0; b < Bb; ++b) { const float d = Y[(size_t)b * D + o] - m; v += d * d; }
  v /= (float)Bb;
  const float inv = rsqrtf(v + EPSBN);
  for (int b = 0; b < Bb; ++b) {
    const float z = (Y[(size_t)b * D + o] - m) * inv * g[o] + be[o];
    Y[(size_t)b * D + o] = fmaxf(z, 0.0f);
  }
}

__global__ void zero_f32(float* p, int n)
{
  const int i = blockIdx.x * blockDim.x + threadIdx.x;
  if (i < n) p[i] = 0.0f;
}

// =====================================================================
// Host orchestration
// =====================================================================
struct LayerP { const float* W; const float* g; const float* be; int out; };

static int parse_mlp(void* const* d_in, int p, const int* outs, int n, LayerP* L)
{
  for (int j = 0; j < n; ++j) {
    L[j].W  = (const float*)d_in[p++];
    p++;                                   // bias: cancels inside batch-norm
    L[j].g  = (const float*)d_in[p++];
    L[j].be = (const float*)d_in[p++];
    L[j].out = outs[j];
  }
  return p;
}

static void run_scale(hipStream_t stream,
                      const float* xyz, const float* pts,
                      int B, int N, int S, int Kk, int Cpts,
                      const LayerP* layers, int nlay,
                      float* bufA, float* bufB, float* wpad, int* idxbuf,
                      float* bnsc, float* bnsh,
                      float* Pout, int Ctot, int coff)
{
  knn_topk<<<B * S, TKT, 0, stream>>>(xyz, B, N, S, Kk, idxbuf);

  const size_t cols = (size_t)B * S * Kk;
  const int Cin = 3 + Cpts;
  const int Kp  = (Cin + 3) & ~3;          // pad first-layer K to multiple of 4
  {
    const unsigned nb = (unsigned)((cols + 255) / 256);
    build_feat<<<nb, 256, 0, stream>>>(xyz, pts, idxbuf, bufA, B, N, S, Kk, Cpts, Kp);
  }
  pad_weights<<<(layers[0].out * Kp + 255) / 256, 256, 0, stream>>>(
      layers[0].W, wpad, layers[0].out, Cin, Kp);

  float* cur = bufA;
  float* nxt = bufB;
  int inC = Cin;
  for (int li = 0; li < nlay; ++li) {
    const int M = layers[li].out;
    const float* Wuse = (li == 0) ? wpad : layers[li].W;
    const int    Kuse = (li == 0) ? Kp   : inC;      // inC multiple of 16 for li>0
    dim3 gg((unsigned)(cols / 256), (unsigned)(M / 16));
    gemm_wmma_f32<<<gg, 128, 0, stream>>>(Wuse, cur, nxt, Kuse, (int)cols);
    bn_stats<<<M, 256, 0, stream>>>(nxt, cols, layers[li].g, layers[li].be, bnsc, bnsh);
    dim3 ga((unsigned)((cols + 255) / 256), (unsigned)M);
    bn_apply<<<ga, 256, 0, stream>>>(nxt, cols, bnsc, bnsh, (li < nlay - 1) ? 1 : 0);
    float* t = cur; cur = nxt; nxt = t;
    inC = M;
  }
  const int tot = B * inC * S;
  maxpool_k<<<(tot + 255) / 256, 256, 0, stream>>>(cur, Pout, B, S, Kk, inC, Ctot, coff);
}

extern "C" void kernel_launch(void* const* d_in, const int* in_sizes, int n_in,
                              void* d_out, int out_size, void* d_ws, size_t ws_size,
                              hipStream_t stream)
{
  const int B = 8, Npts = 4096;

  // ---- parse inputs (top-level insertion order; nested fc dict sorted) ----
  int p = 0;
  const float* x = (const float*)d_in[p++];

  static const int m1a[3] = {32, 32, 64};
  static const int m1b[3] = {64, 64, 128};
  static const int m1c[3] = {64, 96, 128};
  static const int m2a[3] = {64, 64, 128};
  static const int m2b[3] = {128, 128, 256};
  static const int m2c[3] = {128, 128, 256};
  static const int m3a[3] = {256, 512, 1024};

  LayerP L1[3][3], L2[3][3], L3[3];
  p = parse_mlp(d_in, p, m1a, 3, L1[0]);
  p = parse_mlp(d_in, p, m1b, 3, L1[1]);
  p = parse_mlp(d_in, p, m1c, 3, L1[2]);
  p = parse_mlp(d_in, p, m2a, 3, L2[0]);
  p = parse_mlp(d_in, p, m2b, 3, L2[1]);
  p = parse_mlp(d_in, p, m2c, 3, L2[2]);
  p = parse_mlp(d_in, p, m3a, 3, L3);
  // fc_params dict flattened with sorted keys: bn1, bn2, fc1, fc2, fc3
  const float* g1  = (const float*)d_in[p++];
  const float* be1 = (const float*)d_in[p++];
  const float* g2  = (const float*)d_in[p++];
  const float* be2 = (const float*)d_in[p++];
  const float* W1  = (const float*)d_in[p++];
  p++; // fc1 bias cancels in bn1
  const float* W2  = (const float*)d_in[p++];
  p++; // fc2 bias cancels in bn2
  const float* W3  = (const float*)d_in[p++];
  const float* b3  = (const float*)d_in[p++];
  (void)n_in; (void)in_sizes;

  // ---- workspace layout ----
  size_t off = 0;
  float* wsf = (float*)d_ws;
  auto alloc = [&](size_t n) { float* r = wsf + off; off += n; return r; };
  float* xyzA = alloc((size_t)B * 3 * Npts);
  float* ptsA = alloc((size_t)B * 3 * Npts);
  float* xyz1 = alloc((size_t)B * 3 * 512);
  float* pts1 = alloc((size_t)B * 320 * 512);
  float* xyz2 = alloc((size_t)B * 3 * 128);
  float* pts2 = alloc((size_t)B * 640 * 128);
  float* pts3 = alloc((size_t)B * 1024 * 128);
  float* h1   = alloc((size_t)B * 1024);
  float* h2   = alloc((size_t)B * 512);
  float* h3   = alloc((size_t)B * 256);
  float* bnsc = alloc(1024);
  float* bnsh = alloc(1024);
  float* wpad = alloc((size_t)256 * 644);     // padded first-layer weights (max 256x644)
  int*   idxb = (int*)alloc((size_t)B * 512 * 128);
  float* bufA = alloc((size_t)96  * 524288);  // gathered feat / even layer outputs
  float* bufB = alloc((size_t)128 * 524288);  // odd layer outputs
  const size_t need = off * sizeof(float);
  if (ws_size < need) {   // cannot run without scratch; emit zeros deterministically
    zero_f32<<<(out_size + 255) / 256, 256, 0, stream>>>((float*)d_out, out_size);
    return;
  }

  // ---- split x into xyz / points ----
  slice_ch<<<(B * 3 * Npts + 255) / 256, 256, 0, stream>>>(x, xyzA, B, 6, Npts, 0, 3);
  slice_ch<<<(B * 3 * Npts + 255) / 256, 256, 0, stream>>>(x, ptsA, B, 6, Npts, 3, 3);

  // ---- SA1: N=4096 -> S=512, scales K=16/32/128, out channels 64+128+128=320 ----
  run_scale(stream, xyzA, ptsA, B, Npts, 512, 16,  3, L1[0], 3, bufA, bufB, wpad, idxb, bnsc, bnsh, pts1, 320, 0);
  run_scale(stream, xyzA, ptsA, B, Npts, 512, 32,  3, L1[1], 3, bufA, bufB, wpad, idxb, bnsc, bnsh, pts1, 320, 64);
  run_scale(stream, xyzA, ptsA, B, Npts, 512, 128, 3, L1[2], 3, bufA, bufB, wpad, idxb, bnsc, bnsh, pts1, 320, 192);
  slice_n<<<(B * 3 * 512 + 255) / 256, 256, 0, stream>>>(xyzA, xyz1, B, 3, Npts, 512);

  // ---- SA2: N=512 -> S=128, K=32/64/128, out 128+256+256=640 ----
  run_scale(stream, xyz1, pts1, B, 512, 128, 32,  320, L2[0], 3, bufA, bufB, wpad, idxb, bnsc, bnsh, pts2, 640, 0);
  run_scale(stream, xyz1, pts1, B, 512, 128, 64,  320, L2[1], 3, bufA, bufB, wpad, idxb, bnsc, bnsh, pts2, 640, 128);
  run_scale(stream, xyz1, pts1, B, 512, 128, 128, 320, L2[2], 3, bufA, bufB, wpad, idxb, bnsc, bnsh, pts2, 640, 384);
  slice_n<<<(B * 3 * 128 + 255) / 256, 256, 0, stream>>>(xyz1, xyz2, B, 3, 512, 128);

  // ---- SA3: S=N=128, K=1, out 1024 ----
  run_scale(stream, xyz2, pts2, B, 128, 128, 1, 640, L3, 3, bufA, bufB, wpad, idxb, bnsc, bnsh, pts3, 1024, 0);

  // ---- head ----
  max_over_s<<<(B * 1024 + 255) / 256, 256, 0, stream>>>(pts3, h1, B, 1024, 128);
  fc_dot<<<(B * 512 + 255) / 256, 256, 0, stream>>>(h1, W1, nullptr, h2, B, 1024, 512);
  bn_batch<<<(512 + 255) / 256, 256, 0, stream>>>(h2, B, 512, g1, be1);
  fc_dot<<<(B * 256 + 255) / 256, 256, 0, stream>>>(h2, W2, nullptr, h3, B, 512, 256);
  bn_batch<<<(256 + 255) / 256, 256, 0, stream>>>(h3, B, 256, g2, be2);
  fc_dot<<<(B * 40 + 255) / 256, 256, 0, stream>>>(h3, W3, b3, (float*)d_out, B, 256, 40);
}